// GCN_33243046871768
// MI455X (gfx1250) — compile-verified
//
#include <hip/hip_runtime.h>

typedef float v2f __attribute__((ext_vector_type(2)));
typedef float v8f __attribute__((ext_vector_type(8)));

// ---------------------------------------------------------------------------
// Utility kernels
// ---------------------------------------------------------------------------

__global__ void gcn_copy4(float4* __restrict__ dst, const float4* __restrict__ src, int n4) {
    int i = blockIdx.x * blockDim.x + threadIdx.x;
    if (i < n4) dst[i] = src[i];
}

__global__ void gcn_zero4(float4* __restrict__ dst, int n4) {
    int i = blockIdx.x * blockDim.x + threadIdx.x;
    if (i < n4) dst[i] = make_float4(0.f, 0.f, 0.f, 0.f);
}

// Pack W (row-major [128 out][128 in], nn.Linear convention) into exact
// B-fragment order for V_WMMA_F32_16X16X4_F32:
//   B is W^T tile (K x N). Fragment layout (wave32):
//     lane l, vgpr v:  k = kbase + (l>>4)*2 + v,  n = l & 15
//   Packed index = ((ntile*32 + kstep)*32 + lane)*2 + v   -> coalesced b64 loads.
__global__ void gcn_pack_w(const float* __restrict__ W, float* __restrict__ Wp) {
    int idx = blockIdx.x * blockDim.x + threadIdx.x;
    if (idx >= 128 * 128) return;
    int v     = idx & 1;
    int lane  = (idx >> 1) & 31;
    int kstep = (idx >> 6) & 31;
    int ntile = idx >> 11;
    int n = ntile * 16 + (lane & 15);
    int k = kstep * 4 + ((lane >> 4) << 1) + v;
    Wp[idx] = W[n * 128 + k];  // B[k][n] = W[n][k]
}

// ---------------------------------------------------------------------------
// Degree / normalization (edge_index is layer-invariant: compute once)
// ---------------------------------------------------------------------------

__global__ void gcn_degree(const int* __restrict__ ei, float* __restrict__ deg, int E) {
    int e = blockIdx.x * blockDim.x + threadIdx.x;
    if (e >= E) return;
    int r = ei[e];
    int c = ei[E + e];
    if (r != c) atomicAdd(&deg[r], 1.0f);
}

__global__ void gcn_dinv(float* __restrict__ deg, int n) {
    int i = blockIdx.x * blockDim.x + threadIdx.x;
    if (i >= n) return;
    float d = deg[i];
    deg[i] = (d > 0.f) ? rsqrtf(d) : 0.f;
}

// ---------------------------------------------------------------------------
// WMMA fp32 GEMM:  Hlin = H @ W^T + b
//   One wave per 16-node M-tile, all 8 N-tiles (16x128 output per wave).
//   K-loop: 32 steps of K=4  ->  256 V_WMMA_F32_16X16X4_F32 per wave.
//   Full-tile fast path: nNodes = 50000 is a multiple of 16, so the guarded
//   partial-tile path is cold; the hot epilogue is 64 straight b32 stores.
// ---------------------------------------------------------------------------

__global__ void __launch_bounds__(256)
gcn_gemm_wmma(const float* __restrict__ H, const float* __restrict__ Wp,
              const float* __restrict__ bias, float* __restrict__ out, int nNodes) {
    const int lane  = threadIdx.x & 31;
    const int wave  = threadIdx.x >> 5;
    const int mtile = blockIdx.x * 8 + wave;
    if (mtile * 16 >= nNodes) return;  // wave-uniform: EXEC stays all-ones for WMMA

    const int  mBase    = mtile * 16;
    const bool fullTile = (mBase + 16 <= nNodes);  // wave-uniform
    const int  mRow     = mBase + (lane & 15);
    const int  mClmp    = (mRow < nNodes) ? mRow : (nNodes - 1);
    const int  kHalf    = (lane >> 4) << 1;  // 0 or 2

    v8f acc[8];
#pragma unroll
    for (int t = 0; t < 8; ++t) acc[t] = (v8f)(0.f);

    const float* __restrict__ hrow = H + (size_t)mClmp * 128;

    for (int kstep = 0; kstep < 32; ++kstep) {
        const int k = kstep * 4 + kHalf;
        v2f a;
        a.x = hrow[k];
        a.y = hrow[k + 1];
#pragma unroll
        for (int nt = 0; nt < 8; ++nt) {
            const v2f b = *(const v2f*)(Wp + (((nt * 32 + kstep) * 32 + lane) << 1));
            // D = A(16x4 f32) x B(4x16 f32) + C, full fp32
            acc[nt] = __builtin_amdgcn_wmma_f32_16x16x4_f32(
                false, a, false, b, (short)0, acc[nt], false, false);
        }
    }

    // C/D layout: vgpr v -> M = mBase + (lane>=16 ? 8 : 0) + v,  N = lane & 15
    const int nLocal = lane & 15;
    const int mOff   = (lane >> 4) * 8;
    float* __restrict__ outBase = out + (size_t)(mBase + mOff) * 128 + nLocal;

    if (fullTile) {
        // Hot path: unconditional stores, no exec manipulation.
#pragma unroll
        for (int nt = 0; nt < 8; ++nt) {
            const float bv = bias[nt * 16 + nLocal];
#pragma unroll
            for (int v = 0; v < 8; ++v) {
                outBase[(size_t)v * 128 + nt * 16] = acc[nt][v] + bv;
            }
        }
    } else {
        // Cold path: last partial tile only.
#pragma unroll
        for (int nt = 0; nt < 8; ++nt) {
            const float bv = bias[nt * 16 + nLocal];
#pragma unroll
            for (int v = 0; v < 8; ++v) {
                const int m = mBase + mOff + v;
                if (m < nNodes) out[(size_t)m * 128 + nt * 16 + nLocal] = acc[nt][v] + bv;
            }
        }
    }
}

// ---------------------------------------------------------------------------
// Edge aggregation: agg[col] += dinv[row]*dinv[col] * Hlin[row]
//   One wave32 per edge; each lane handles a float4 slice of the 128-dim msg.
// ---------------------------------------------------------------------------

__global__ void __launch_bounds__(256)
gcn_aggregate(const float* __restrict__ Hlin, const int* __restrict__ ei,
              const float* __restrict__ dinv, float* __restrict__ agg, int E) {
    const int lane = threadIdx.x & 31;
    const int wave = threadIdx.x >> 5;
    const int e    = blockIdx.x * 8 + wave;
    if (e >= E) return;
    const int r = ei[e];
    const int c = ei[E + e];
    if (r == c) return;  // remove_self_loops
    const float norm = dinv[r] * dinv[c];

    const float4 m = *((const float4*)(Hlin + (size_t)r * 128) + lane);
    float* dst = agg + (size_t)c * 128 + lane * 4;
    atomicAdd(dst + 0, norm * m.x);
    atomicAdd(dst + 1, norm * m.y);
    atomicAdd(dst + 2, norm * m.z);
    atomicAdd(dst + 3, norm * m.w);
}

// ---------------------------------------------------------------------------
// Residual + ReLU:  h = h + relu(agg)
// ---------------------------------------------------------------------------

__global__ void gcn_residual_relu(float4* __restrict__ h, const float4* __restrict__ agg, int n4) {
    int i = blockIdx.x * blockDim.x + threadIdx.x;
    if (i >= n4) return;
    float4 hv = h[i];
    float4 av = agg[i];
    hv.x += fmaxf(av.x, 0.f);
    hv.y += fmaxf(av.y, 0.f);
    hv.z += fmaxf(av.z, 0.f);
    hv.w += fmaxf(av.w, 0.f);
    h[i] = hv;
}

// ---------------------------------------------------------------------------
// Launch
// ---------------------------------------------------------------------------

extern "C" void kernel_launch(void* const* d_in, const int* in_sizes, int n_in,
                              void* d_out, int out_size, void* d_ws, size_t ws_size,
                              hipStream_t stream) {
    const float* x  = (const float*)d_in[0];
    const int*   ei = (const int*)d_in[1];
    const float* W[3] = {(const float*)d_in[2], (const float*)d_in[4], (const float*)d_in[6]};
    const float* b[3] = {(const float*)d_in[3], (const float*)d_in[5], (const float*)d_in[7]};

    const int nNodes = in_sizes[0] / 128;   // 50000
    const int E      = in_sizes[1] / 2;     // 800000
    const int nFeat  = nNodes * 128;
    const int nFeat4 = nFeat / 4;

    float* H = (float*)d_out;  // persistent node features (starts as x)

    // Workspace layout (floats)
    float* ws   = (float*)d_ws;
    float* Wp   = ws;                       // 3 * 16384 packed weights
    float* dinv = Wp + 3 * 16384;           // nNodes (deg, then rsqrt in place)
    float* Hlin = dinv + ((nNodes + 3) & ~3);
    float* agg  = Hlin + nFeat;

    // h = x
    gcn_copy4<<<(nFeat4 + 255) / 256, 256, 0, stream>>>((float4*)H, (const float4*)x, nFeat4);

    // degrees -> dinv (once; edge structure is the same for all layers)
    gcn_zero4<<<((nNodes + 3) / 4 + 255) / 256, 256, 0, stream>>>((float4*)dinv, (nNodes + 3) / 4);
    gcn_degree<<<(E + 255) / 256, 256, 0, stream>>>(ei, dinv, E);
    gcn_dinv<<<(nNodes + 255) / 256, 256, 0, stream>>>(dinv, nNodes);

    // pack all three weight matrices into WMMA B-fragment order
    for (int l = 0; l < 3; ++l)
        gcn_pack_w<<<(128 * 128 + 255) / 256, 256, 0, stream>>>(W[l], Wp + l * 16384);

    const int mTiles     = (nNodes + 15) / 16;
    const int gemmBlocks = (mTiles + 7) / 8;
    const int aggBlocks  = (E + 7) / 8;

    for (int l = 0; l < 3; ++l) {
        gcn_gemm_wmma<<<gemmBlocks, 256, 0, stream>>>(H, Wp + l * 16384, b[l], Hlin, nNodes);
        gcn_zero4<<<(nFeat4 + 255) / 256, 256, 0, stream>>>((float4*)agg, nFeat4);
        gcn_aggregate<<<aggBlocks, 256, 0, stream>>>(Hlin, ei, dinv, agg, E);
        gcn_residual_relu<<<(nFeat4 + 255) / 256, 256, 0, stream>>>((float4*)H, (const float4*)agg, nFeat4);
    }
}